// VectorizationGenerator_30210799960520
// MI455X (gfx1250) — compile-verified
//
#include <hip/hip_runtime.h>
#include <math.h>

#define SEQ 2048
#define NCH 8
#define NAR 6
#define TPB 256
#define KPT (SEQ / TPB)   // 8 timesteps per thread

typedef __attribute__((ext_vector_type(2))) float v2f;
typedef __attribute__((ext_vector_type(8))) float v8f;

#if defined(__gfx1250__)
#if __has_builtin(__builtin_amdgcn_wmma_f32_16x16x4_f32)
#define USE_WMMA_F32 1
#endif
#endif

__device__ __forceinline__ float block_reduce_sum(float v, float* s_red) {
  const int tid = threadIdx.x;
  __syncthreads();
  s_red[tid] = v;
  __syncthreads();
  for (int off = TPB >> 1; off > 0; off >>= 1) {
    if (tid < off) s_red[tid] += s_red[tid + off];
    __syncthreads();
  }
  return s_red[0];
}

// One block per batch row. Produces partials[16][nb] (rows 9..15 zero).
__global__ __launch_bounds__(TPB) void vg_row_kernel(
    const float* __restrict__ ytc,   // y_true_command [B,S,8]
    const float* __restrict__ ytx,   // y_true_coord   [B,S,6]
    const float* __restrict__ ypc,   // y_pred_command [B,S,8]
    const float* __restrict__ ypx,   // y_pred_coord   [B,S,6]
    float* __restrict__ partials,    // [16][nb]
    int nb)
{
  const int b   = blockIdx.x;
  const int tid = threadIdx.x;

  __shared__ float  s_psoc[SEQ];          // softmax prob of SOC per t
  __shared__ float  s_peos[SEQ];          // softmax prob of EOS per t
  __shared__ float  s_nll[SEQ];           // -log_softmax[true_idx] per t
  __shared__ float2 s_xyT[SEQ];           // true xy (valid t only)
  __shared__ float2 s_xyP[SEQ];           // pred xy (valid t only)
  __shared__ unsigned char s_idx[SEQ];    // true command index per t
  __shared__ int   s_teos, s_peosv;
  __shared__ float s_red[TPB];

  if (tid == 0) { s_teos = SEQ; s_peosv = SEQ; }
  __syncthreads();

  const float* tc_base = ytc + (size_t)b * SEQ * NCH;
  const float* pc_base = ypc + (size_t)b * SEQ * NCH;

  // ---------------- Phase A: command tensors, softmax, eos scan -----------
  int   loc_teos = SEQ, loc_peos = SEQ;
  float loc_handle = 0.f;

  for (int k = 0; k < KPT; ++k) {
    const int t = tid + k * TPB;
    if (k + 1 < KPT) {
      __builtin_prefetch(tc_base + (size_t)(t + TPB) * NCH, 0, 0);
      __builtin_prefetch(pc_base + (size_t)(t + TPB) * NCH, 0, 0);
    }
    const float4* tc4 = (const float4*)(tc_base + (size_t)t * NCH);
    const float4  t0 = tc4[0], t1 = tc4[1];
    const float tv[8] = {t0.x, t0.y, t0.z, t0.w, t1.x, t1.y, t1.z, t1.w};
    int ti = 0; float bv = tv[0];
#pragma unroll
    for (int c = 1; c < 8; ++c) if (tv[c] > bv) { bv = tv[c]; ti = c; }
    s_idx[t] = (unsigned char)ti;
    if (ti == 7 && t < loc_teos) loc_teos = t;      // EOS == 7

    const float4* pc4 = (const float4*)(pc_base + (size_t)t * NCH);
    const float4  p0 = pc4[0], p1 = pc4[1];
    const float pv[8] = {p0.x, p0.y, p0.z, p0.w, p1.x, p1.y, p1.z, p1.w};
    int pi = 0; float pm = pv[0];
#pragma unroll
    for (int c = 1; c < 8; ++c) if (pv[c] > pm) { pm = pv[c]; pi = c; }
    if (pi == 7 && t < loc_peos) loc_peos = t;

    float e[8], den = 0.f;
#pragma unroll
    for (int c = 0; c < 8; ++c) { e[c] = expf(pv[c] - pm); den += e[c]; }
    const float inv = 1.0f / den;
    s_psoc[t] = e[0] * inv;                  // SOC == 0
    s_peos[t] = e[7] * inv;                  // EOS == 7
    s_nll[t]  = logf(den) + pm - pv[ti];
    loc_handle += (e[1] - e[2] - e[3]) * inv;    // N_ - NH - NV, unmasked
  }
  atomicMin(&s_teos, loc_teos);
  atomicMin(&s_peosv, loc_peos);
  __syncthreads();

  const int teos   = s_teos;
  const int peos   = s_peosv;
  const int maxlen = (teos > peos) ? teos : peos;

  // ---------------- Phase B: coords, masked sums --------------------------
  const int ac[8] = {0, 6, 4, 4, 2, 1, 1, 0};    // ARG_COUNTS
  float loc_cnum = 0.f, loc_cden = 0.f, loc_cmd = 0.f;
  float loc_soc  = 0.f, loc_socc = 0.f, loc_len = 0.f;

  for (int k = 0; k < KPT; ++k) {
    const int t = tid + k * TPB;
    if (t < maxlen) loc_cmd += s_nll[t];
    if (t < teos) {
      const float2* tr2 = (const float2*)(ytx + ((size_t)b * SEQ + t) * NAR);
      const float2* pr2 = (const float2*)(ypx + ((size_t)b * SEQ + t) * NAR);
      const float2 ta = tr2[0], tb = tr2[1], tc = tr2[2];
      const float2 pa = pr2[0], pb = pr2[1], pc = pr2[2];
      const float trv[6] = {ta.x, ta.y, tb.x, tb.y, tc.x, tc.y};
      const float prv[6] = {pa.x, pa.y, pb.x, pb.y, pc.x, pc.y};
      s_xyT[t] = make_float2(trv[0], trv[1]);
      s_xyP[t] = make_float2(prv[0], prv[1]);
      const int na = ac[s_idx[t]];
      for (int j = 0; j < na; ++j) {
        const float d = prv[j] - trv[j];
        const float ad = fabsf(d);
        loc_cnum += (ad <= 1.0f) ? 0.5f * d * d : (ad - 0.5f);   // Huber, delta=1
      }
      loc_cden += (float)na;
      loc_soc  += s_psoc[t];
      loc_len  += 1.0f - s_peos[t];
      if (s_idx[t] == 0) loc_socc += 1.0f;
    }
  }
  __syncthreads();   // s_xyT/s_xyP visible to all before segment scan

  // ---------------- Phase C: per-segment shoelace areas -------------------
  float loc_sa = 0.f, loc_nc = 0.f;
  for (int k = 0; k < KPT; ++k) {
    const int t = tid + k * TPB;
    if (t < teos && (s_idx[t] == 0 || t == 0)) {   // segment start
      int e2 = t + 1;
      while (e2 < teos && s_idx[e2] != 0) ++e2;    // next start or eos
      const int len = e2 - t;
      loc_nc += 1.0f;                               // len >= 1 always
      if (len >= 3) {
        const float2 fT = s_xyT[t], fP = s_xyP[t];
        float aT = 0.f, aP = 0.f;
        for (int u = t; u < e2; ++u) {
          const float2 uT = s_xyT[u], uP = s_xyP[u];
          const float2 nT = (u == e2 - 1) ? fT : s_xyT[u + 1];
          const float2 nP = (u == e2 - 1) ? fP : s_xyP[u + 1];
          aT += uT.x * nT.y - uT.y * nT.x;
          aP += uP.x * nP.y - uP.y * nP.x;
        }
        loc_sa += fabsf(0.5f * aT - 0.5f * aP);
      }
    }
  }

  // ---------------- Reduce & emit per-row partials ------------------------
  const float r_cnum   = block_reduce_sum(loc_cnum,   s_red);
  const float r_cden   = block_reduce_sum(loc_cden,   s_red);
  const float r_cmd    = block_reduce_sum(loc_cmd,    s_red);
  const float r_soc    = block_reduce_sum(loc_soc,    s_red);
  const float r_socc   = block_reduce_sum(loc_socc,   s_red);
  const float r_len    = block_reduce_sum(loc_len,    s_red);
  const float r_handle = block_reduce_sum(loc_handle, s_red);
  const float r_sa     = block_reduce_sum(loc_sa,     s_red);
  const float r_nc     = block_reduce_sum(loc_nc,     s_red);

  if (tid == 0) {
    partials[0 * nb + b] = r_cnum;
    partials[1 * nb + b] = r_cden;
    partials[2 * nb + b] = r_cmd;
    partials[3 * nb + b] = (float)maxlen;                 // sum(mask2) per row
    partials[4 * nb + b] = fabsf(r_soc - r_socc);         // contour term
    partials[5 * nb + b] = fabsf(r_len - (float)teos);    // eos term
    partials[6 * nb + b] = r_handle;
    partials[7 * nb + b] = r_sa;
    partials[8 * nb + b] = r_nc;
    for (int q = 9; q < 16; ++q) partials[q * nb + b] = 0.f;
  }
}

// Single wave32. Cross-row reduction via V_WMMA_F32_16X16X4_F32:
// D(16x16) = A(16x4) * ones(4x16) + C  =>  D[m][0] = sum over columns of row m.
__global__ __launch_bounds__(32) void vg_finalize_kernel(
    const float* __restrict__ partials, float* __restrict__ out, int nb)
{
  __shared__ float s_sum[16];
  const int l = threadIdx.x;
#if defined(USE_WMMA_F32)
  v8f c = {};
  const v2f bones = {1.0f, 1.0f};
  const int m  = l & 15;     // A-matrix row (ISA 16x4 f32 layout)
  const int kh = l >> 4;     // lanes 16-31 hold K=2,3
  const int iters = nb >> 2; // 4 columns per WMMA
  for (int i = 0; i < iters; ++i) {
    v2f a;
    const int col = i * 4 + kh * 2;
    a.x = partials[m * nb + col];       // VGPR0: K = 0 (or 2)
    a.y = partials[m * nb + col + 1];   // VGPR1: K = 1 (or 3)
    c = __builtin_amdgcn_wmma_f32_16x16x4_f32(
        /*neg_a=*/false, a, /*neg_b=*/false, bones,
        /*c_mod=*/(short)0, c, /*reuse_a=*/false, /*reuse_b=*/false);
  }
  // D[m][0]: m=0..7 -> lane 0 c[m]; m=8..15 -> lane 16 c[m-8]
  if (l == 0)  { for (int j = 0; j < 8; ++j) s_sum[j]     = c[j]; }
  if (l == 16) { for (int j = 0; j < 8; ++j) s_sum[8 + j] = c[j]; }
#else
  if (l < 16) {
    float s = 0.f;
    for (int r = 0; r < nb; ++r) s += partials[l * nb + r];
    s_sum[l] = s;
  }
#endif
  __syncthreads();
  if (l == 0) {
    const float fB = (float)nb;
    out[0] = s_sum[0] / s_sum[1];                       // coord_loss
    out[1] = s_sum[2] / (s_sum[3] + 1e-8f);             // cmd_loss
    out[2] = s_sum[4] / fB;                             // contour_loss
    out[3] = s_sum[5] / fB;                             // eos_loss
    out[4] = s_sum[6] / (fB * (float)SEQ);              // handle_loss
    out[5] = s_sum[7] / fmaxf(s_sum[8], 1.0f);          // sa_loss
  }
}

extern "C" void kernel_launch(void* const* d_in, const int* in_sizes, int n_in,
                              void* d_out, int out_size, void* d_ws, size_t ws_size,
                              hipStream_t stream) {
  const float* ytc = (const float*)d_in[0];   // y_true_command [B,S,8]
  const float* ytx = (const float*)d_in[1];   // y_true_coord   [B,S,6]
  const float* ypc = (const float*)d_in[2];   // y_pred_command [B,S,8]
  const float* ypx = (const float*)d_in[3];   // y_pred_coord   [B,S,6]
  float* out      = (float*)d_out;
  float* partials = (float*)d_ws;             // needs 16*nb*4 bytes (32 KB)

  const int nb = in_sizes[0] / (SEQ * NCH);   // B = 512

  vg_row_kernel<<<nb, TPB, 0, stream>>>(ytc, ytx, ypc, ypx, partials, nb);
  vg_finalize_kernel<<<1, 32, 0, stream>>>(partials, out, nb);
}